// LRENet_adv_34342558499514
// MI455X (gfx1250) — compile-verified
//
#include <hip/hip_runtime.h>
#include <math.h>

// ---------------------------------------------------------------------------
// LRENet forward for gfx1250 (MI455X). fp32 end-to-end; all linear layers go
// through a WMMA f32 16x16x4 GEMM. Each wave computes a 16x64 output strip
// (4 independent accumulators). The K-loop is explicitly software-pipelined,
// and the GEMM is declared __launch_bounds__(32,1) so the backend allocates
// registers for latency hiding instead of multi-wave occupancy (several of
// this model's GEMMs launch only ~64 waves on the whole GPU).
// ---------------------------------------------------------------------------

typedef __attribute__((ext_vector_type(2))) float v2f;
typedef __attribute__((ext_vector_type(8))) float v8f;

#define EMAX   9
#define CAP    29
#define EROWS  (EMAX*CAP)       // 261
#define EPAD   272              // padded to multiple of 16
#define NTOK   256
#define DMODEL 512
#define NSRC   1024
#define GN     4                // 16-wide N tiles per wave (strip = 64)

#if defined(__gfx1250__) && __has_builtin(__builtin_amdgcn_wmma_f32_16x16x4_f32)
#define HAVE_WMMA_F32 1
#else
#define HAVE_WMMA_F32 0
#endif

// Y[M][N] = X[M][K] @ W[N][K]^T + bias[N] (+ residual[M][N])
// grid = (N/64, ceil(M/16)), block = 32 (one wave per 16x64 strip).
__global__ __launch_bounds__(32, 1)
void gemm_wmma(const float* __restrict__ X, int ldx,
               const float* __restrict__ W, int ldw,
               const float* __restrict__ bias,
               const float* __restrict__ residual, int ldr,
               float* __restrict__ Y, int ldy,
               int M, int N, int K) {
  const int m0 = blockIdx.y * 16;
  const int n0 = blockIdx.x * (16 * GN);
  const int lane = threadIdx.x & 31;
  const int half = lane >> 4;      // selects K pair {0,1} vs {2,3}
  const int lrow = lane & 15;
  v8f acc[GN] = {};
#if HAVE_WMMA_F32
  // Lane-dependent K offset (2*half) folded into the stream base pointers.
  const float* xp  = X + (size_t)(m0 + lrow) * ldx + 2 * half;           // A: M = lrow
  const float* wp0 = W + (size_t)(n0 + 0 * 16 + lrow) * ldw + 2 * half;  // B(=W^T): N = lrow + 16j
  const float* wp1 = W + (size_t)(n0 + 1 * 16 + lrow) * ldw + 2 * half;
  const float* wp2 = W + (size_t)(n0 + 2 * 16 + lrow) * ldw + 2 * half;
  const float* wp3 = W + (size_t)(n0 + 3 * 16 + lrow) * ldw + 2 * half;
  // Software pipeline, depth 1: prologue loads step 0.
  v2f a0, b0, b1, b2, b3;
  a0.x = xp[0];  a0.y = xp[1];
  b0.x = wp0[0]; b0.y = wp0[1];
  b1.x = wp1[0]; b1.y = wp1[1];
  b2.x = wp2[0]; b2.y = wp2[1];
  b3.x = wp3[0]; b3.y = wp3[1];
#pragma unroll 4
  for (int k = 0; k + 4 < K; k += 4) {
    // Issue next step's loads before consuming the current fragments.
    const int kn = k + 4;
    v2f an, bn0, bn1, bn2, bn3;
    an.x  = xp[kn];  an.y  = xp[kn + 1];
    bn0.x = wp0[kn]; bn0.y = wp0[kn + 1];
    bn1.x = wp1[kn]; bn1.y = wp1[kn + 1];
    bn2.x = wp2[kn]; bn2.y = wp2[kn + 1];
    bn3.x = wp3[kn]; bn3.y = wp3[kn + 1];
    acc[0] = __builtin_amdgcn_wmma_f32_16x16x4_f32(false, a0, false, b0, (short)0, acc[0], false, false);
    acc[1] = __builtin_amdgcn_wmma_f32_16x16x4_f32(false, a0, false, b1, (short)0, acc[1], false, false);
    acc[2] = __builtin_amdgcn_wmma_f32_16x16x4_f32(false, a0, false, b2, (short)0, acc[2], false, false);
    acc[3] = __builtin_amdgcn_wmma_f32_16x16x4_f32(false, a0, false, b3, (short)0, acc[3], false, false);
    a0 = an; b0 = bn0; b1 = bn1; b2 = bn2; b3 = bn3;   // renamed away under unroll
  }
  // Epilogue: last step.
  acc[0] = __builtin_amdgcn_wmma_f32_16x16x4_f32(false, a0, false, b0, (short)0, acc[0], false, false);
  acc[1] = __builtin_amdgcn_wmma_f32_16x16x4_f32(false, a0, false, b1, (short)0, acc[1], false, false);
  acc[2] = __builtin_amdgcn_wmma_f32_16x16x4_f32(false, a0, false, b2, (short)0, acc[2], false, false);
  acc[3] = __builtin_amdgcn_wmma_f32_16x16x4_f32(false, a0, false, b3, (short)0, acc[3], false, false);
#else
  // scalar fallback with identical output mapping (host pass / missing builtin)
  for (int j = 0; j < GN; ++j)
    for (int v = 0; v < 8; ++v) {
      const int row = m0 + v + 8 * half;
      float s = 0.f;
      for (int k = 0; k < K; ++k)
        s += X[(size_t)row * ldx + k] * W[(size_t)(n0 + 16 * j + lrow) * ldw + k];
      acc[j][v] = s;
    }
#endif
  // C/D layout: lane -> N, VGPR v -> M = v + 8*half
#pragma unroll
  for (int j = 0; j < GN; ++j) {
    const int col = n0 + 16 * j + lrow;
    const float bv = bias ? bias[col] : 0.f;
    for (int v = 0; v < 8; ++v) {
      const int row = m0 + v + 8 * half;
      if (row < M) {
        float val = acc[j][v] + bv;
        if (residual) val += residual[(size_t)row * ldr + col];
        Y[(size_t)row * ldy + col] = val;
      }
    }
  }
}

// LayerNorm over `cols` (optionally ReLU). One block per row, 256 threads.
__global__ void ln_kernel(const float* __restrict__ x, int ldx, int xoff,
                          const float* __restrict__ g, const float* __restrict__ b,
                          float* __restrict__ y, int ldy, int cols, int relu) {
  __shared__ float sm[256];
  const int row = blockIdx.x, t = threadIdx.x, nt = blockDim.x;
  const float* xr = x + (size_t)row * ldx + xoff;
  float* yr = y + (size_t)row * ldy;
  float s = 0.f;
  for (int c = t; c < cols; c += nt) s += xr[c];
  sm[t] = s; __syncthreads();
  for (int k = nt >> 1; k > 0; k >>= 1) { if (t < k) sm[t] += sm[t + k]; __syncthreads(); }
  const float mean = sm[0] / cols; __syncthreads();
  float v = 0.f;
  for (int c = t; c < cols; c += nt) { float d = xr[c] - mean; v += d * d; }
  sm[t] = v; __syncthreads();
  for (int k = nt >> 1; k > 0; k >>= 1) { if (t < k) sm[t] += sm[t + k]; __syncthreads(); }
  const float inv = rsqrtf(sm[0] / cols + 1e-5f);
  for (int c = t; c < cols; c += nt) {
    float o = (xr[c] - mean) * inv * g[c] + b[c];
    if (relu) o = fmaxf(o, 0.f);
    yr[c] = o;
  }
}

// qkf attention: nheads = Dm, head_dim = 1. One block per feature h; thread t
// handles query l=t. K/V column h staged in LDS.
__global__ void attn_hd1(const float* __restrict__ Q, const float* __restrict__ Km,
                         const float* __restrict__ Vm, float* __restrict__ O) {
  __shared__ float ks[NSRC];
  __shared__ float vs[NSRC];
  const int h = blockIdx.x;
  for (int s = threadIdx.x; s < NSRC; s += blockDim.x) {
    ks[s] = Km[(size_t)s * DMODEL + h];
    vs[s] = Vm[(size_t)s * DMODEL + h];
  }
  __syncthreads();
  const int l = threadIdx.x;
  const float q = Q[(size_t)l * DMODEL + h];
  float m = -1e30f;
  for (int s = 0; s < NSRC; ++s) m = fmaxf(m, q * ks[s]);
  float sum = 0.f, accv = 0.f;
  for (int s = 0; s < NSRC; ++s) {
    const float e = __expf(q * ks[s] - m);
    sum += e; accv += e * vs[s];
  }
  O[(size_t)l * DMODEL + h] = accv / sum;
}

// Block MHA: QKV packed (L x 3*Dm), heads of hd dims. grid (heads, L), 256 thr.
__global__ void attn_mh(const float* __restrict__ QKV, float* __restrict__ O,
                        int Dm, int hd, float scale) {
  __shared__ float w[256];
  __shared__ float red[256];
  const int h = blockIdx.x, l = blockIdx.y, t = threadIdx.x;
  const float* qrow = QKV + (size_t)l * 3 * Dm + h * hd;
  const float* krow = QKV + (size_t)t * 3 * Dm + Dm + h * hd;
  float dot = 0.f;
  for (int d = 0; d < hd; ++d) dot += qrow[d] * krow[d];
  dot *= scale;
  red[t] = dot; __syncthreads();
  for (int s = 128; s > 0; s >>= 1) { if (t < s) red[t] = fmaxf(red[t], red[t + s]); __syncthreads(); }
  const float m = red[0]; __syncthreads();
  const float e = __expf(dot - m);
  red[t] = e; __syncthreads();
  for (int s = 128; s > 0; s >>= 1) { if (t < s) red[t] += red[t + s]; __syncthreads(); }
  const float sum = red[0]; __syncthreads();
  w[t] = e / sum; __syncthreads();
  if (t < hd) {
    float acc = 0.f;
    for (int s = 0; s < NTOK; ++s) acc += w[s] * QKV[(size_t)s * 3 * Dm + 2 * Dm + h * hd + t];
    O[(size_t)l * Dm + h * hd + t] = acc;
  }
}

// proj row L2-normalize (in place). block per row.
__global__ void rownorm_kernel(float* __restrict__ x, int cols) {
  __shared__ float sm[256];
  const int row = blockIdx.x, t = threadIdx.x, nt = blockDim.x;
  float s = 0.f;
  for (int c = t; c < cols; c += nt) { float v = x[(size_t)row * cols + c]; s += v * v; }
  sm[t] = s; __syncthreads();
  for (int k = nt >> 1; k > 0; k >>= 1) { if (t < k) sm[t] += sm[t + k]; __syncthreads(); }
  const float inv = 1.f / fmaxf(sqrtf(sm[0]), 1e-12f);
  for (int c = t; c < cols; c += nt) x[(size_t)row * cols + c] *= inv;
}

// sim column normalize (256 x 9).
__global__ void simnorm_kernel(const float* __restrict__ sim, float* __restrict__ simn, int rows) {
  const int e = threadIdx.x;
  if (e >= EMAX) return;
  float s = 0.f;
  for (int r = 0; r < rows; ++r) { float v = sim[r * EMAX + e]; s += v * v; }
  const float inv = 1.f / fmaxf(sqrtf(s), 1e-12f);
  for (int r = 0; r < rows; ++r) simn[r * EMAX + e] = sim[r * EMAX + e] * inv;
}

// logits = proj_n @ sim_n * exp(min(temp, log 100)). grid (NTOK), block 16.
__global__ void gate_logits_kernel(const float* __restrict__ projn,
                                   const float* __restrict__ simn,
                                   const float* __restrict__ temp,
                                   float* __restrict__ lg, int Kd) {
  const int t = blockIdx.x, e = threadIdx.x;
  if (e >= EMAX) return;
  const float scale = __expf(fminf(temp[0], 4.6051702f));
  float acc = 0.f;
  for (int k = 0; k < Kd; ++k) acc += projn[(size_t)t * Kd + k] * simn[k * EMAX + e];
  lg[t * EMAX + e] = acc * scale;
}

// Softmax gate, argmax, importance/load aux losses (deterministic, no atomics).
__global__ void gate_kernel(const float* __restrict__ lg, float* __restrict__ gatev,
                            int* __restrict__ idxv, float* __restrict__ aux, float sigma) {
  __shared__ float sc_s[NTOK * EMAX];
  __shared__ float thr_s[NTOK];
  __shared__ float red[2 * EMAX];
  const int t = threadIdx.x;
  float lr[EMAX], sc[EMAX];
  float mx = -1e30f; int am = 0;
  for (int e = 0; e < EMAX; ++e) { lr[e] = lg[t * EMAX + e]; if (lr[e] > mx) { mx = lr[e]; am = e; } }
  float sum = 0.f;
  for (int e = 0; e < EMAX; ++e) { sc[e] = __expf(lr[e] - mx); sum += sc[e]; }
  const float inv = 1.f / sum;
  for (int e = 0; e < EMAX; ++e) { sc[e] *= inv; sc_s[t * EMAX + e] = sc[e]; }
  thr_s[t] = lr[am];
  gatev[t] = sc[am];
  idxv[t] = am;
  __syncthreads();
  if (t < EMAX) {
    const float den = sigma * 1.41421356237f;
    float imp = 0.f, ld = 0.f;
    for (int r = 0; r < NTOK; ++r) {
      const float s = sc_s[r * EMAX + t];
      imp += s;
      ld += 0.5f * (1.f + erff((s - thr_s[r]) / den));
    }
    red[t] = imp; red[EMAX + t] = ld;
  }
  __syncthreads();
  if (t == 0) {
    float mi = 0.f, ml = 0.f;
    for (int e = 0; e < EMAX; ++e) { mi += red[e]; ml += red[EMAX + e]; }
    mi /= EMAX; ml /= EMAX;
    float li = 0.f, ll = 0.f;
    for (int e = 0; e < EMAX; ++e) {
      float a = red[e] - mi; li += a * a;
      float b = red[EMAX + e] - ml; ll += b * b;
    }
    li = li / (EMAX - 1) / (mi * mi + 1e-10f);
    ll = ll / (EMAX - 1) / (ml * ml + 1e-10f);
    aux[0] += 0.01f * 0.5f * (li + ll);
  }
}

// Stable descending sort by gate + per-expert rank + capacity drop.
__global__ void route_kernel(const float* __restrict__ gatev, const int* __restrict__ idxv,
                             int* __restrict__ tloc, int* __restrict__ ssrc) {
  __shared__ float g[NTOK];
  __shared__ int id[NTOK];
  __shared__ int ord[NTOK];
  __shared__ int ids[NTOK];
  const int t = threadIdx.x;
  g[t] = gatev[t]; id[t] = idxv[t];
  for (int r = t; r < EPAD; r += NTOK) ssrc[r] = -1;
  __syncthreads();
  const float gt = g[t];
  int pos = 0;
  for (int u = 0; u < NTOK; ++u) {
    const float gu = g[u];
    if (gu > gt || (gu == gt && u < t)) pos++;
  }
  ord[pos] = t;
  __syncthreads();
  ids[t] = id[ord[t]];
  __syncthreads();
  const int my = ids[t];
  int rank = 0;
  for (int j = 0; j < t; ++j) if (ids[j] == my) rank++;
  if (rank < CAP) {
    const int loc = my * CAP + rank;
    ssrc[loc] = ord[t];
    tloc[ord[t]] = loc;
  } else {
    tloc[ord[t]] = -1;
  }
}

__global__ void scatter_kernel(const float* __restrict__ x, const int* __restrict__ ssrc,
                               float* __restrict__ buf, int rows_valid, int cols) {
  const int r = blockIdx.x;
  const int src = (r < rows_valid) ? ssrc[r] : -1;
  for (int c = threadIdx.x; c < cols; c += blockDim.x)
    buf[(size_t)r * cols + c] = (src >= 0) ? x[(size_t)src * cols + c] : 0.f;
}

__global__ void gather_kernel(const float* __restrict__ y, const int* __restrict__ tloc,
                              const float* __restrict__ gatev, float* __restrict__ outp, int cols) {
  const int t = blockIdx.x;
  const int loc = tloc[t];
  const float gv = gatev[t];
  for (int c = threadIdx.x; c < cols; c += blockDim.x)
    outp[(size_t)t * cols + c] = (loc >= 0) ? y[(size_t)loc * cols + c] * gv : 0.f;
}

// x += rowmean(x) (in place). block per row.
__global__ void addmean_kernel(float* __restrict__ x, int cols) {
  __shared__ float sm[256];
  const int row = blockIdx.x, t = threadIdx.x, nt = blockDim.x;
  float s = 0.f;
  for (int c = t; c < cols; c += nt) s += x[(size_t)row * cols + c];
  sm[t] = s; __syncthreads();
  for (int k = nt >> 1; k > 0; k >>= 1) { if (t < k) sm[t] += sm[t + k]; __syncthreads(); }
  const float m = sm[0] / cols;
  for (int c = t; c < cols; c += nt) x[(size_t)row * cols + c] += m;
}

__global__ void combine_kernel(const float* __restrict__ f1, const float* __restrict__ f2,
                               float* __restrict__ cur, int n) {
  const int i = blockIdx.x * blockDim.x + threadIdx.x;
  if (i < n) cur[i] = 0.25f * (f1[i] + f1[n + i] + f2[i] + f2[n + i]);
}

__global__ void gelu_kernel(float* __restrict__ x, int n) {
  const int i = blockIdx.x * blockDim.x + threadIdx.x;
  if (i < n) { const float h = x[i]; x[i] = h / (1.f + __expf(-1.702f * h)); }
}

__global__ void zerof_kernel(float* __restrict__ p, int n) {
  const int i = blockIdx.x * blockDim.x + threadIdx.x;
  if (i < n) p[i] = 0.f;
}

// Final head: logits = hh_flat @ Wc^T + bc; hazards, argmax, aux pack.
__global__ void head_kernel(const float* __restrict__ hh, const float* __restrict__ Wc,
                            const float* __restrict__ bc, const float* __restrict__ aux,
                            float* __restrict__ out, int Kf) {
  __shared__ float r0[256];
  __shared__ float r1[256];
  const int t = threadIdx.x;
  float a = 0.f, b = 0.f;
  for (int k = t; k < Kf; k += 256) { const float h = hh[k]; a += h * Wc[k]; b += h * Wc[Kf + k]; }
  r0[t] = a; r1[t] = b; __syncthreads();
  for (int s = 128; s > 0; s >>= 1) { if (t < s) { r0[t] += r0[t + s]; r1[t] += r1[t + s]; } __syncthreads(); }
  if (t == 0) {
    const float l0 = r0[0] + bc[0], l1 = r1[0] + bc[1];
    out[0] = l0; out[1] = l1;
    out[2] = 1.f / (1.f + __expf(-l0));
    out[3] = 1.f / (1.f + __expf(-l1));
    out[4] = (l1 > l0) ? 1.f : 0.f;
    out[5] = aux[0];
  }
}

// ---------------------------------------------------------------------------

static inline void gemm(hipStream_t s, const float* X, int ldx, const float* W, int ldw,
                        const float* bias, const float* resid, int ldr,
                        float* Y, int ldy, int M, int N, int K) {
  dim3 g((unsigned)(N / (16 * GN)), (unsigned)((M + 15) / 16)), b(32);
  gemm_wmma<<<g, b, 0, s>>>(X, ldx, W, ldw, bias, resid, ldr, Y, ldy, M, N, K);
}

extern "C" void kernel_launch(void* const* d_in, const int* in_sizes, int n_in,
                              void* d_out, int out_size, void* d_ws, size_t ws_size,
                              hipStream_t stream) {
  (void)in_sizes; (void)n_in; (void)out_size; (void)ws_size;
  int p = 0;
  auto F = [&](void) { return (const float*)d_in[p++]; };

  // --- inputs in setup_inputs() dict (insertion) order -----------------------
  const float* share = F();
  const float* tokens[2]; tokens[0] = F(); tokens[1] = F();
  struct IndP { const float *ln1g, *ln1b, *ln2g, *ln2b, *Win, *bin, *Wout, *bout, *Wm, *bm; } ind[2];
  for (int i = 0; i < 2; ++i) {
    ind[i].ln1g = F(); ind[i].ln1b = F(); ind[i].ln2g = F(); ind[i].ln2b = F();
    ind[i].Win = F(); ind[i].bin = F(); ind[i].Wout = F(); ind[i].bout = F();
    ind[i].Wm = F(); ind[i].bm = F();
  }
  const float *Wp = F(), *bp = F(), *sim = F(), *temp = F();
  const float *W1 = F(), *b1 = F(), *g1 = F(), *be1 = F();
  const float *W2 = F(), *b2 = F(), *g2 = F(), *be2 = F();
  const float *W3 = F(), *b3 = F(), *g3 = F(), *be3 = F();
  const float *W4 = F(), *b4 = F();
  const float *bln1g = F(), *bln1b = F(), *bln2g = F(), *bln2b = F();
  const float *bWin = F(), *bbin = F(), *bWout = F(), *bbout = F();
  const float *bWfc = F(), *bbfc = F(), *bWpr = F(), *bbpr = F();
  const float *Wb = F(), *bb = F(), *Wc = F(), *bc = F();

  float* out = (float*)d_out;
  float* f1o = out + 6;                       // (2, 256, 512)
  float* f2o = out + 6 + 2 * NTOK * DMODEL;   // (2, 256, 512)

  // --- workspace -------------------------------------------------------------
  float* ws = (float*)d_ws;
  size_t o = 0;
  auto A = [&](size_t n) { float* q = ws + o; o += n; return q; };
  float* sfln  = A((size_t)NSRC * DMODEL);
  float* tkln  = A((size_t)NTOK * DMODEL);
  float* Qb    = A((size_t)NTOK * DMODEL);
  float* Kb    = A((size_t)NSRC * DMODEL);
  float* Vb    = A((size_t)NSRC * DMODEL);
  float* Ob    = A((size_t)NTOK * DMODEL);
  float* valb  = A((size_t)NTOK * DMODEL);
  float* projb = A((size_t)NTOK * 256);
  float* simn  = A((size_t)256 * EMAX);
  float* lgb   = A((size_t)NTOK * EMAX);
  float* gatev = A(NTOK);
  int*   idxv  = (int*)A(NTOK);
  int*   tloc  = (int*)A(NTOK);
  int*   ssrc  = (int*)A(EPAD + 16);
  float* auxp  = A(8);
  float* bufb  = A((size_t)EPAD * DMODEL);
  float* ebuf  = A((size_t)EPAD * DMODEL);
  float* e2buf = A((size_t)EPAD * 256);
  float* dbuf  = A((size_t)EPAD * DMODEL);
  float* ybuf  = A((size_t)EPAD * DMODEL);
  float* curb  = A((size_t)NTOK * DMODEL);
  float* ylnb  = A((size_t)NTOK * DMODEL);
  float* qkvb  = A((size_t)NTOK * 3 * DMODEL);
  float* attnb = A((size_t)NTOK * DMODEL);
  float* xnb   = A((size_t)NTOK * DMODEL);
  float* hqb   = A((size_t)NTOK * 4 * DMODEL);
  float* xob   = A((size_t)NTOK * DMODEL);
  float* hhb   = A((size_t)NTOK * 256);

  const float sigma = 1.0f / (float)EMAX;

  zerof_kernel<<<1, 32, 0, stream>>>(auxp, 1);

  for (int i = 0; i < 2; ++i) {
    const int off = i * DMODEL;
    const IndP& ip = ind[i];
    float* q_i = f1o + (size_t)i * NTOK * DMODEL;

    // qkf: LN + projections + hd=1 attention + Wout + Wm
    ln_kernel<<<NSRC, 256, 0, stream>>>(share, 2 * DMODEL, off, ip.ln1g, ip.ln1b, sfln, DMODEL, DMODEL, 0);
    ln_kernel<<<NTOK, 256, 0, stream>>>(tokens[i], DMODEL, 0, ip.ln2g, ip.ln2b, tkln, DMODEL, DMODEL, 0);
    gemm(stream, tkln, DMODEL, ip.Win, DMODEL, ip.bin, nullptr, 0, Qb, DMODEL, NTOK, DMODEL, DMODEL);
    gemm(stream, sfln, DMODEL, ip.Win + (size_t)DMODEL * DMODEL, DMODEL, ip.bin + DMODEL, nullptr, 0,
         Kb, DMODEL, NSRC, DMODEL, DMODEL);
    gemm(stream, sfln, DMODEL, ip.Win + (size_t)2 * DMODEL * DMODEL, DMODEL, ip.bin + 2 * DMODEL, nullptr, 0,
         Vb, DMODEL, NSRC, DMODEL, DMODEL);
    attn_hd1<<<DMODEL, 256, 0, stream>>>(Qb, Kb, Vb, Ob);
    gemm(stream, Ob, DMODEL, ip.Wout, DMODEL, ip.bout, nullptr, 0, valb, DMODEL, NTOK, DMODEL, DMODEL);
    gemm(stream, valb, DMODEL, ip.Wm, DMODEL, ip.bm, nullptr, 0, q_i, DMODEL, NTOK, DMODEL, DMODEL);

    // MoE gating
    gemm(stream, q_i, DMODEL, Wp, DMODEL, bp, nullptr, 0, projb, 256, NTOK, 256, DMODEL);
    rownorm_kernel<<<NTOK, 256, 0, stream>>>(projb, 256);
    simnorm_kernel<<<1, 16, 0, stream>>>(sim, simn, 256);
    gate_logits_kernel<<<NTOK, 16, 0, stream>>>(projb, simn, temp, lgb, 256);
    gate_kernel<<<1, NTOK, 0, stream>>>(lgb, gatev, idxv, auxp, sigma);
    route_kernel<<<1, NTOK, 0, stream>>>(gatev, idxv, tloc, ssrc);
    scatter_kernel<<<EPAD, 256, 0, stream>>>(q_i, ssrc, bufb, EROWS, DMODEL);

    // Expert MLP (padded to 272 rows)
    gemm(stream, bufb, DMODEL, W1, DMODEL, b1, nullptr, 0, ebuf, DMODEL, EPAD, DMODEL, DMODEL);
    ln_kernel<<<EPAD, 256, 0, stream>>>(ebuf, DMODEL, 0, g1, be1, ebuf, DMODEL, DMODEL, 1);
    gemm(stream, ebuf, DMODEL, W2, DMODEL, b2, nullptr, 0, e2buf, 256, EPAD, 256, DMODEL);
    ln_kernel<<<EPAD, 256, 0, stream>>>(e2buf, 256, 0, g2, be2, e2buf, 256, 256, 1);
    addmean_kernel<<<EPAD, 256, 0, stream>>>(e2buf, 256);
    gemm(stream, e2buf, 256, W3, 256, b3, nullptr, 0, dbuf, DMODEL, EPAD, DMODEL, 256);
    ln_kernel<<<EPAD, 256, 0, stream>>>(dbuf, DMODEL, 0, g3, be3, dbuf, DMODEL, DMODEL, 1);
    gemm(stream, dbuf, DMODEL, W4, DMODEL, b4, nullptr, 0, ybuf, DMODEL, EPAD, DMODEL, DMODEL);
    gather_kernel<<<NTOK, 256, 0, stream>>>(ybuf, tloc, gatev,
                                            f2o + (size_t)i * NTOK * DMODEL, DMODEL);
  }

  // Transformer block
  combine_kernel<<<(NTOK * DMODEL + 255) / 256, 256, 0, stream>>>(f1o, f2o, curb, NTOK * DMODEL);
  ln_kernel<<<NTOK, 256, 0, stream>>>(curb, DMODEL, 0, bln1g, bln1b, ylnb, DMODEL, DMODEL, 0);
  gemm(stream, ylnb, DMODEL, bWin, DMODEL, bbin, nullptr, 0, qkvb, 3 * DMODEL, NTOK, 3 * DMODEL, DMODEL);
  attn_mh<<<dim3(8, NTOK), 256, 0, stream>>>(qkvb, attnb, DMODEL, 64, 0.125f);
  gemm(stream, attnb, DMODEL, bWout, DMODEL, bbout, curb, DMODEL, xnb, DMODEL, NTOK, DMODEL, DMODEL);
  ln_kernel<<<NTOK, 256, 0, stream>>>(xnb, DMODEL, 0, bln2g, bln2b, ylnb, DMODEL, DMODEL, 0);
  gemm(stream, ylnb, DMODEL, bWfc, DMODEL, bbfc, nullptr, 0, hqb, 4 * DMODEL, NTOK, 4 * DMODEL, DMODEL);
  gelu_kernel<<<(NTOK * 4 * DMODEL + 255) / 256, 256, 0, stream>>>(hqb, NTOK * 4 * DMODEL);
  gemm(stream, hqb, 4 * DMODEL, bWpr, 4 * DMODEL, bbpr, xnb, DMODEL, xob, DMODEL, NTOK, DMODEL, 4 * DMODEL);
  gemm(stream, xob, DMODEL, Wb, DMODEL, bb, nullptr, 0, hhb, 256, NTOK, 256, DMODEL);
  head_kernel<<<1, 256, 0, stream>>>(hhb, Wc, bc, auxp, out, NTOK * 256);
}